// GCNModel_79568564126006
// MI455X (gfx1250) — compile-verified
//
#include <hip/hip_runtime.h>

#define N_NODES 100000
#define N_EDGES 1600000
#define N_GRAPH 512
#define F_INP   74
#define HDIM    256
#define F_OUT   12

typedef __attribute__((ext_vector_type(2))) float v2f;
typedef __attribute__((ext_vector_type(8))) float v8f;

// ---------------------------------------------------------------- utilities
__global__ void k_fill(float* __restrict__ p, float v, int n) {
    int i = blockIdx.x * blockDim.x + threadIdx.x;
    if (i < n) p[i] = v;
}

// deg[dst] += 1  (deg buffer pre-filled with 1.0 for the self loop)
__global__ void k_deg(const int* __restrict__ ei, float* __restrict__ deg, int E) {
    int e = blockIdx.x * blockDim.x + threadIdx.x;
    if (e < E) unsafeAtomicAdd(&deg[ei[E + e]], 1.0f);
}

__global__ void k_rsqrt(float* __restrict__ d, int n) {
    int i = blockIdx.x * blockDim.x + threadIdx.x;
    if (i < n) d[i] = rsqrtf(d[i]);
}

// -------------------------------------------------- fp32 WMMA GEMM  C = A*W
// one wave -> one 16x16 output tile via V_WMMA_F32_16X16X4_F32
// block = 4 waves covering 16 rows x 64 cols; grid = (N/16, HDIM/64)
__global__ __launch_bounds__(128) void k_gemm(const float* __restrict__ A,
                                              const float* __restrict__ W,
                                              float* __restrict__ C,
                                              int K, int ldA) {
    const int lane = threadIdx.x & 31;
    const int wave = threadIdx.x >> 5;
    const int row0 = blockIdx.x * 16;
    const int col0 = blockIdx.y * 64 + wave * 16;
    const int m  = lane & 15;          // row within A-tile / col within B-tile
    const int kx = (lane >> 4) << 1;   // 0 or 2: which K pair this half-wave owns

    const float* __restrict__ Arow = A + (size_t)(row0 + m) * ldA;
    v8f acc = {};

    const int Kfloor = K & ~3;
    #pragma unroll 4
    for (int k = 0; k < Kfloor; k += 4) {
        const int k0 = k + kx;
        v2f a, b;
        a.x = Arow[k0];
        a.y = Arow[k0 + 1];
        b.x = W[(size_t)k0 * HDIM + col0 + m];
        b.y = W[(size_t)(k0 + 1) * HDIM + col0 + m];
        acc = __builtin_amdgcn_wmma_f32_16x16x4_f32(false, a, false, b,
                                                    (short)0, acc, false, false);
    }
    if (K & 3) {                       // predicated 4-wide tail (zero-fill)
        const int k0 = Kfloor + kx, k1 = k0 + 1;
        v2f a, b;
        a.x = (k0 < K) ? Arow[k0] : 0.f;
        a.y = (k1 < K) ? Arow[k1] : 0.f;
        b.x = (k0 < K) ? W[(size_t)k0 * HDIM + col0 + m] : 0.f;
        b.y = (k1 < K) ? W[(size_t)k1 * HDIM + col0 + m] : 0.f;
        acc = __builtin_amdgcn_wmma_f32_16x16x4_f32(false, a, false, b,
                                                    (short)0, acc, false, false);
    }

    // D layout: VGPR r -> M = r (lanes 0-15) / r+8 (lanes 16-31), N = lane&15
    const int nc    = col0 + m;
    const int rbase = row0 + ((lane >> 4) << 3);
    #pragma unroll
    for (int r = 0; r < 8; ++r)
        C[(size_t)(rbase + r) * HDIM + nc] = acc[r];
}

// ------------------------------------------ acc = t * dis[i]^2 + bias  (self loop + bias)
__global__ void k_selfbias(const float* __restrict__ t, const float* __restrict__ dis,
                           const float* __restrict__ bias, float* __restrict__ acc) {
    int idx = blockIdx.x * blockDim.x + threadIdx.x;     // < N*HDIM
    int i = idx >> 8, f = idx & 255;
    float d = dis[i];
    acc[idx] = t[idx] * d * d + bias[f];
}

// ------------------------------------------ edge scatter: acc[dst] += t[src]*norm
// 64 threads per edge, float4 per thread, no-return fp32 atomics
__global__ void k_scatter(const int* __restrict__ ei, const float* __restrict__ dis,
                          const float* __restrict__ t, float* __restrict__ acc, int E) {
    int tid = blockIdx.x * blockDim.x + threadIdx.x;
    int e = tid >> 6;
    if (e >= E) return;
    int c = (tid & 63) << 2;
    int s = ei[e];
    int d = ei[E + e];
    float nrm = dis[s] * dis[d];
    const float4 v = *reinterpret_cast<const float4*>(t + (size_t)s * HDIM + c);
    float* p = acc + (size_t)d * HDIM + c;
    unsafeAtomicAdd(p + 0, v.x * nrm);
    unsafeAtomicAdd(p + 1, v.y * nrm);
    unsafeAtomicAdd(p + 2, v.z * nrm);
    unsafeAtomicAdd(p + 3, v.w * nrm);
}

__global__ void k_relu(float* __restrict__ p, int n) {
    int i = blockIdx.x * blockDim.x + threadIdx.x;
    if (i < n) p[i] = fmaxf(p[i], 0.0f);
}

// ------------------------------------------ pooling
__global__ void k_count(const int* __restrict__ batch, float* __restrict__ cnt, int n) {
    int i = blockIdx.x * blockDim.x + threadIdx.x;
    if (i < n) unsafeAtomicAdd(&cnt[batch[i]], 1.0f);
}

__global__ void k_pool(const int* __restrict__ batch, const float* __restrict__ h,
                       float* __restrict__ pooled, int n) {
    int tid = blockIdx.x * blockDim.x + threadIdx.x;     // n * 64 threads
    int i = tid >> 6;
    if (i >= n) return;
    int c = (tid & 63) << 2;
    int g = batch[i];
    const float4 v = *reinterpret_cast<const float4*>(h + (size_t)i * HDIM + c);
    float* p = pooled + (size_t)g * HDIM + c;
    unsafeAtomicAdd(p + 0, v.x);
    unsafeAtomicAdd(p + 1, v.y);
    unsafeAtomicAdd(p + 2, v.z);
    unsafeAtomicAdd(p + 3, v.w);
}

__global__ void k_fc(const float* __restrict__ pooled, const float* __restrict__ cnt,
                     const float* __restrict__ Wfc, const float* __restrict__ bfc,
                     float* __restrict__ out) {
    int idx = blockIdx.x * blockDim.x + threadIdx.x;     // < G * F_OUT
    if (idx >= N_GRAPH * F_OUT) return;
    int g = idx / F_OUT, o = idx - g * F_OUT;
    float inv = 1.0f / fmaxf(cnt[g], 1.0f);
    const float* pr = pooled + (size_t)g * HDIM;
    float s = 0.f;
    #pragma unroll 8
    for (int f = 0; f < HDIM; ++f) s += pr[f] * Wfc[f * F_OUT + o];
    out[idx] = s * inv + bfc[o];
}

// ---------------------------------------------------------------- launcher
extern "C" void kernel_launch(void* const* d_in, const int* in_sizes, int n_in,
                              void* d_out, int out_size, void* d_ws, size_t ws_size,
                              hipStream_t stream) {
    const float* x    = (const float*)d_in[0];
    const int*   ei   = (const int*)d_in[1];
    const int*   batch= (const int*)d_in[2];
    const float* W1   = (const float*)d_in[3];
    const float* b1   = (const float*)d_in[4];
    const float* W2   = (const float*)d_in[5];
    const float* b2   = (const float*)d_in[6];
    const float* W3   = (const float*)d_in[7];
    const float* b3   = (const float*)d_in[8];
    const float* Wfc  = (const float*)d_in[9];
    const float* bfc  = (const float*)d_in[10];
    float* out = (float*)d_out;

    float* dis    = (float*)d_ws;                       // N (deg -> deg^-1/2)
    float* A      = dis + N_NODES;                      // N*H : h / acc (ping)
    float* B      = A + (size_t)N_NODES * HDIM;         // N*H : t = h@W
    float* pooled = B + (size_t)N_NODES * HDIM;         // G*H
    float* cnt    = pooled + (size_t)N_GRAPH * HDIM;    // G

    const int T = 256;
    const int nhB = (N_NODES * HDIM) / T;               // 100000 blocks
    const int scB = (N_EDGES * 64) / T;                 // 400000 blocks
    const dim3 gg(N_NODES / 16, HDIM / 64);             // (6250, 4)

    // symmetric normalization:  dis = (1 + in-degree)^-1/2
    k_fill <<<(N_NODES + T - 1) / T, T, 0, stream>>>(dis, 1.0f, N_NODES);
    k_deg  <<<(N_EDGES + T - 1) / T, T, 0, stream>>>(ei, dis, N_EDGES);
    k_rsqrt<<<(N_NODES + T - 1) / T, T, 0, stream>>>(dis, N_NODES);

    // layer 1 (K = 74)
    k_gemm    <<<gg, 128, 0, stream>>>(x, W1, B, F_INP, F_INP);
    k_selfbias<<<nhB, T, 0, stream>>>(B, dis, b1, A);
    k_scatter <<<scB, T, 0, stream>>>(ei, dis, B, A, N_EDGES);
    k_relu    <<<nhB, T, 0, stream>>>(A, N_NODES * HDIM);

    // layer 2
    k_gemm    <<<gg, 128, 0, stream>>>(A, W2, B, HDIM, HDIM);
    k_selfbias<<<nhB, T, 0, stream>>>(B, dis, b2, A);
    k_scatter <<<scB, T, 0, stream>>>(ei, dis, B, A, N_EDGES);
    k_relu    <<<nhB, T, 0, stream>>>(A, N_NODES * HDIM);

    // layer 3
    k_gemm    <<<gg, 128, 0, stream>>>(A, W3, B, HDIM, HDIM);
    k_selfbias<<<nhB, T, 0, stream>>>(B, dis, b3, A);
    k_scatter <<<scB, T, 0, stream>>>(ei, dis, B, A, N_EDGES);
    k_relu    <<<nhB, T, 0, stream>>>(A, N_NODES * HDIM);

    // mean pool + FC
    k_fill <<<(N_GRAPH * HDIM + T - 1) / T, T, 0, stream>>>(pooled, 0.0f, N_GRAPH * HDIM);
    k_fill <<<(N_GRAPH + T - 1) / T, T, 0, stream>>>(cnt, 0.0f, N_GRAPH);
    k_count<<<(N_NODES + T - 1) / T, T, 0, stream>>>(batch, cnt, N_NODES);
    k_pool <<<(N_NODES * 64 + T - 1) / T, T, 0, stream>>>(batch, A, pooled, N_NODES);
    k_fc   <<<(N_GRAPH * F_OUT + T - 1) / T, T, 0, stream>>>(pooled, cnt, Wfc, bfc, out);
}